// TimesBlock_35888746725927
// MI455X (gfx1250) — compile-verified
//
#include <hip/hip_runtime.h>
#include <math.h>

#define SEQ_T 480
#define NCH   16
#define NB    8
#define NF    241   // 480/2 + 1
#define NTOT  (NB * SEQ_T * NCH)

typedef float v2f __attribute__((ext_vector_type(2)));
typedef float v8f __attribute__((ext_vector_type(8)));

// ---- workspace layout (float offsets, 16-aligned sections) ----
#define OFF_AMP    0                 // [B][241][16]  = 30848
#define OFF_AMPF   30848             // [241][16]     = 3856   (batch-mean, f=0 zeroed)
#define OFF_PWSRC  34704             // [B][241]      = 1928 (pad 1936)  channel-mean amp
#define OFF_TOPF   36640             // [2][16] ints  = 32
#define OFF_PW     36672             // [B][2][16]    = 256   softmax weights
#define OFF_W1M    36928             // [32][25]      = 800   merged conv1 (1 in-ch)
#define OFF_B1M    37728             // [32]
#define OFF_W2M    37760             // [16][800]     = 12800 merged conv2, [co][tap*32+ci]
#define OFF_B2M    50560             // [16] (pad to 50592)
#define OFF_RES    50592             // [2][B][480][16] = 122880

// ------------------------------------------------------------------
// 1) Merge inception weights: mean over 3 kernels folded into one 5x5.
//    conv1 additionally sums over its 16 identical input channels.
//    W2m stored tap-major: [co][tap*32 + ci] so WMMA K-chunks stay
//    inside one tap (32 % 4 == 0) and operands are contiguous.
// ------------------------------------------------------------------
__global__ void tb_prep(const float* __restrict__ w1_0, const float* __restrict__ b1_0,
                        const float* __restrict__ w1_1, const float* __restrict__ b1_1,
                        const float* __restrict__ w1_2, const float* __restrict__ b1_2,
                        const float* __restrict__ w2_0, const float* __restrict__ b2_0,
                        const float* __restrict__ w2_1, const float* __restrict__ b2_1,
                        const float* __restrict__ w2_2, const float* __restrict__ b2_2,
                        float* __restrict__ ws) {
  const int idx = blockIdx.x * blockDim.x + threadIdx.x;
  const float third = 1.0f / 3.0f;
  if (idx < 800) {                                  // W1m: [co=32][tap=25]
    const int co = idx / 25, tap = idx % 25;
    const int dy = tap / 5, dx = tap % 5;
    float s = 0.0f;
    for (int ci = 0; ci < 16; ++ci) {
      float v = w1_2[((co * 16 + ci) * 5 + dy) * 5 + dx];
      if (dy >= 1 && dy <= 3 && dx >= 1 && dx <= 3)
        v += w1_1[((co * 16 + ci) * 3 + (dy - 1)) * 3 + (dx - 1)];
      if (dy == 2 && dx == 2) v += w1_0[co * 16 + ci];
      s += v;
    }
    ws[OFF_W1M + idx] = s * third;
  } else if (idx < 832) {                           // b1m
    const int co = idx - 800;
    ws[OFF_B1M + co] = (b1_0[co] + b1_1[co] + b1_2[co]) * third;
  } else if (idx < 832 + 12800) {                   // W2m
    const int r = idx - 832;
    const int co = r / 800, rem = r % 800;
    const int ci = rem / 25, tap = rem % 25;
    const int dy = tap / 5, dx = tap % 5;
    float v = w2_2[((co * 32 + ci) * 5 + dy) * 5 + dx];
    if (dy >= 1 && dy <= 3 && dx >= 1 && dx <= 3)
      v += w2_1[((co * 32 + ci) * 3 + (dy - 1)) * 3 + (dx - 1)];
    if (dy == 2 && dx == 2) v += w2_0[co * 32 + ci];
    ws[OFF_W2M + co * 800 + tap * 32 + ci] = v * third;   // tap-major
  } else if (idx < 832 + 12800 + 16) {              // b2m
    const int co = idx - 832 - 12800;
    ws[OFF_B2M + co] = (b2_0[co] + b2_1[co] + b2_2[co]) * third;
  }
}

// ------------------------------------------------------------------
// 2) |rfft(x, axis=T)| via direct DFT (tiny: 8*241*16 threads x 480)
// ------------------------------------------------------------------
__global__ void tb_dft(const float* __restrict__ x, float* __restrict__ ws) {
  const int idx = blockIdx.x * blockDim.x + threadIdx.x;
  if (idx >= NB * NF * NCH) return;
  const int n = idx % NCH;
  const int f = (idx / NCH) % NF;
  const int b = idx / (NCH * NF);
  const float w = -6.2831853071795864f * (float)f / (float)SEQ_T;
  float cr = 0.0f, cim = 0.0f;
  for (int t = 0; t < SEQ_T; ++t) {
    float s, c;
    __sincosf(w * (float)t, &s, &c);
    const float xv = x[((size_t)b * SEQ_T + t) * NCH + n];
    cr  = fmaf(xv, c, cr);
    cim = fmaf(xv, s, cim);
  }
  ws[OFF_AMP + idx] = sqrtf(cr * cr + cim * cim);
}

// ------------------------------------------------------------------
// 3) Reductions: batch-mean (for top-k, f=0 zeroed) and channel-mean
// ------------------------------------------------------------------
__global__ void tb_reduce(float* __restrict__ ws) {
  const int idx = blockIdx.x * blockDim.x + threadIdx.x;
  const float* amp = ws + OFF_AMP;
  if (idx < NF * NCH) {
    const int f = idx / NCH;
    float s = 0.0f;
    for (int b = 0; b < NB; ++b) s += amp[b * NF * NCH + idx];
    ws[OFF_AMPF + idx] = (f == 0) ? 0.0f : s * (1.0f / NB);
  } else if (idx < NF * NCH + NB * NF) {
    const int r = idx - NF * NCH;
    const int b = r / NF, f = r % NF;
    float s = 0.0f;
    for (int n = 0; n < NCH; ++n) s += amp[(b * NF + f) * NCH + n];
    ws[OFF_PWSRC + r] = s * (1.0f / NCH);
  }
}

// ------------------------------------------------------------------
// 4) Per-channel top-2 frequencies + softmax period weights
// ------------------------------------------------------------------
__global__ __launch_bounds__(128) void tb_topk(float* __restrict__ ws) {
  const int tid = threadIdx.x;
  const float* ampf = ws + OFF_AMPF;
  int* topf = (int*)(ws + OFF_TOPF);
  if (tid < NCH) {
    float v1 = -1e30f, v2 = -1e30f;
    int i1 = 1, i2 = 1;
    for (int f = 0; f < NF; ++f) {
      const float v = ampf[f * NCH + tid];
      if (v > v1)      { v2 = v1; i2 = i1; v1 = v; i1 = f; }
      else if (v > v2) { v2 = v;  i2 = f; }
    }
    topf[tid] = i1;
    topf[NCH + tid] = i2;
  }
  __syncthreads();
  if (tid < NB * NCH) {
    const int b = tid / NCH, n = tid % NCH;
    const float* pws = ws + OFF_PWSRC + b * NF;
    const float w0 = pws[topf[n]];
    const float w1 = pws[topf[NCH + n]];
    const float m  = fmaxf(w0, w1);
    const float e0 = expf(w0 - m), e1 = expf(w1 - m);
    const float inv = 1.0f / (e0 + e1);
    ws[OFF_PW + (b * 2 + 0) * NCH + n] = e0 * inv;
    ws[OFF_PW + (b * 2 + 1) * NCH + n] = e1 * inv;
  }
}

// ------------------------------------------------------------------
// 5) Main kernel: one wave per (k, b, j) image.
//    conv1 (1->32, merged 5x5) in VALU, GELU, then conv2 (32->16,
//    merged 5x5) at the 30 needed pixels via V_WMMA_F32_16X16X4_F32:
//    D[16co x 16pix] += A[16co x 4k] * B[4k x 16pix], K = 25 taps x 32 ci.
// ------------------------------------------------------------------
__global__ __launch_bounds__(32) void tb_conv(const float* __restrict__ x,
                                              float* __restrict__ ws) {
  __shared__ float xs[512];          // folded input image (needed rows)
  __shared__ float h1s[480 * 32];    // gelu(conv1) : [pix][ci]

  const int j = blockIdx.x;          // channel
  const int b = blockIdx.y;          // batch
  const int k = blockIdx.z;          // top-k index
  const int tid = threadIdx.x;

  const int*   topf = (const int*)(ws + OFF_TOPF);
  const float* W1m  = ws + OFF_W1M;
  const float* b1m  = ws + OFF_B1M;
  const float* W2m  = ws + OFF_W2M;
  const float* b2m  = ws + OFF_B2M;
  float* res = ws + OFF_RES + (size_t)(k * NB + b) * SEQ_T * NCH;

  int f = topf[k * NCH + j];
  if (f < 1) f = 1;
  const int p  = SEQ_T / f;                          // period (width)
  const int H  = (SEQ_T + p - 1) / p;                // image height
  const int h_hi = 29 / p;                           // last needed output row
  int rows_hi = h_hi + 2; if (rows_hi > H - 1) rows_hi = H - 1;   // conv1 rows
  int xrows   = rows_hi + 3; if (xrows > H) xrows = H;            // x rows
  int nxpix   = xrows * p;         if (nxpix > 512) nxpix = 512;

  for (int pix = tid; pix < nxpix; pix += 32)        // fold: linear idx == t
    xs[pix] = (pix < SEQ_T) ? x[((size_t)b * SEQ_T + pix) * NCH + j] : 0.0f;
  __syncthreads();

  // conv1 + GELU: lane = output channel (32 lanes = 32 channels)
  float wr[25];
#pragma unroll
  for (int i = 0; i < 25; ++i) wr[i] = W1m[tid * 25 + i];
  const float bias1 = b1m[tid];

  for (int r = 0; r <= rows_hi; ++r) {
    for (int w = 0; w < p; ++w) {
      float acc = bias1;
#pragma unroll
      for (int dy = 0; dy < 5; ++dy) {
        const int row = r + dy - 2;
        if (row < 0 || row >= H) continue;           // uniform across wave
        const int rb = row * p;
#pragma unroll
        for (int dx = 0; dx < 5; ++dx) {
          const int col = w + dx - 2;
          if (col < 0 || col >= p) continue;
          acc = fmaf(wr[dy * 5 + dx], xs[rb + col], acc);
        }
      }
      // exact GELU
      h1s[(r * p + w) * 32 + tid] =
          0.5f * acc * (1.0f + erff(acc * 0.70710678118654752f));
    }
  }
  __syncthreads();

  // conv2 via WMMA f32 16x16x4. Only pixels 0..29 are needed.
  // A lanes 0-15 hold K={0,1} (v0,v1), lanes 16-31 hold K={2,3}; B mirrors
  // with N = lane%16.  K order: tap*32 + ci  ->  chunk never crosses a tap.
  const int lane16 = tid & 15;
  const int lo = (tid < 16) ? 0 : 2;                 // K sub-offset per lane half
  const float* w2row = W2m + lane16 * 800;           // A row M = lane16

  for (int tile = 0; tile < 2; ++tile) {
    const int pixn = tile * 16 + lane16;             // B column's pixel
    const int hq = pixn / p;
    const int wq = pixn - hq * p;
    const int validn = (pixn < 30) ? 1 : 0;

    v8f acc = {0.f, 0.f, 0.f, 0.f, 0.f, 0.f, 0.f, 0.f};
#pragma unroll 1
    for (int tap = 0; tap < 25; ++tap) {
      const int dy  = tap / 5;                       // literal divisors
      const int dx  = tap - dy * 5;
      const int row = hq + dy - 2;
      const int col = wq + dx - 2;
      const int ok  = validn & (int)(row >= 0) & (int)(row < H) &
                      (int)(col >= 0) & (int)(col < p);
      const int idx2 = row * p + col;
      const float* h1p = h1s + (size_t)((ok ? idx2 : 0) * 32) + lo;  // clamped
      const float* w2p = w2row + tap * 32 + lo;
#pragma unroll
      for (int c4 = 0; c4 < 32; c4 += 4) {           // 8 WMMAs per tap
        const float a0 = w2p[c4];
        const float a1 = w2p[c4 + 1];
        float b0 = h1p[c4];
        float b1 = h1p[c4 + 1];
        b0 = ok ? b0 : 0.0f;
        b1 = ok ? b1 : 0.0f;
        v2f A;  A[0] = a0; A[1] = a1;
        v2f Bm; Bm[0] = b0; Bm[1] = b1;
        acc = __builtin_amdgcn_wmma_f32_16x16x4_f32(
            /*neg_a=*/false, A, /*neg_b=*/false, Bm,
            /*c_mod=*/(short)0, acc, /*reuse_a=*/false, /*reuse_b=*/false);
      }
    }

    const int cobase = (tid < 16) ? 0 : 8;           // D layout: vgpr v -> M
    const int pix = tile * 16 + lane16;
#pragma unroll
    for (int v = 0; v < 8; ++v) {
      const int co = cobase + v;
      if (pix < 30) {
        const int t = pix * 16 + co;                 // flatten (h,w,c)[:480]
        res[t * NCH + j] = acc[v] + b2m[co];
      }
    }
  }
}

// ------------------------------------------------------------------
// 6) Combine: out = x + pw0*res0 + pw1*res1
// ------------------------------------------------------------------
__global__ void tb_combine(const float* __restrict__ x, const float* __restrict__ ws,
                           float* __restrict__ out) {
  const int idx = blockIdx.x * blockDim.x + threadIdx.x;
  if (idx >= NTOT) return;
  const int j = idx % NCH;
  const int b = idx / (SEQ_T * NCH);
  const float w0 = ws[OFF_PW + (b * 2 + 0) * NCH + j];
  const float w1 = ws[OFF_PW + (b * 2 + 1) * NCH + j];
  const float r0 = ws[OFF_RES + idx];
  const float r1 = ws[OFF_RES + NTOT + idx];
  out[idx] = x[idx] + w0 * r0 + w1 * r1;
}

extern "C" void kernel_launch(void* const* d_in, const int* in_sizes, int n_in,
                              void* d_out, int out_size, void* d_ws, size_t ws_size,
                              hipStream_t stream) {
  (void)in_sizes; (void)n_in; (void)out_size; (void)ws_size;
  const float* x    = (const float*)d_in[0];
  const float* w1_0 = (const float*)d_in[1];
  const float* b1_0 = (const float*)d_in[2];
  const float* w1_1 = (const float*)d_in[3];
  const float* b1_1 = (const float*)d_in[4];
  const float* w1_2 = (const float*)d_in[5];
  const float* b1_2 = (const float*)d_in[6];
  const float* w2_0 = (const float*)d_in[7];
  const float* b2_0 = (const float*)d_in[8];
  const float* w2_1 = (const float*)d_in[9];
  const float* b2_1 = (const float*)d_in[10];
  const float* w2_2 = (const float*)d_in[11];
  const float* b2_2 = (const float*)d_in[12];
  float* ws  = (float*)d_ws;
  float* out = (float*)d_out;

  tb_prep<<<(832 + 12800 + 16 + 255) / 256, 256, 0, stream>>>(
      w1_0, b1_0, w1_1, b1_1, w1_2, b1_2, w2_0, b2_0, w2_1, b2_1, w2_2, b2_2, ws);
  tb_dft<<<(NB * NF * NCH + 255) / 256, 256, 0, stream>>>(x, ws);
  tb_reduce<<<(NF * NCH + NB * NF + 255) / 256, 256, 0, stream>>>(ws);
  tb_topk<<<1, 128, 0, stream>>>(ws);
  tb_conv<<<dim3(16, NB, 2), 32, 0, stream>>>(x, ws);
  tb_combine<<<(NTOT + 255) / 256, 256, 0, stream>>>(x, ws, out);
}